// SelfAttention_60885456388509
// MI455X (gfx1250) — compile-verified
//
#include <hip/hip_runtime.h>
#include <hip/hip_bf16.h>

typedef float v2f __attribute__((ext_vector_type(2)));
typedef float v8f __attribute__((ext_vector_type(8)));

#define D_DIM 512
#define TL    1024
#define CL    256
#define G_CNT 128
#define ECHUNK 64
#define NEG_SLOPE 0.01f

__device__ __forceinline__ float lrelu(float x) { return x >= 0.f ? x : NEG_SLOPE * x; }

__device__ __forceinline__ v8f wmma_f32(v2f a, v2f b, v8f c) {
  // V_WMMA_F32_16X16X4_F32 : D(16x16 f32) = A(16x4) * B(4x16) + C
  return __builtin_amdgcn_wmma_f32_16x16x4_f32(false, a, false, b, (short)0, c, false, false);
}

// -------- Kernel 1: vsum[g][e] = sum_t mask_total[g][t] * LReLU(<seq_total[g][t], W2[e]>)
// grid (8 e-chunks, 128 groups), 256 threads = 8 waves.
// wave w: e-tile = (w&3) within chunk, half = w>>2 splits the t-tile sequence.
__global__ void vsum_kernel(const float* __restrict__ seq_total,
                            const int*   __restrict__ mask_total,
                            const float* __restrict__ W2,
                            float*       __restrict__ vsum) {
  extern __shared__ __align__(16) float smem[];
  float* Wlds = smem;                                   // 64*512 f32  (128KB)
  float* Alds = Wlds + ECHUNK * D_DIM;                  // 2 tiles of 16*512 (64KB)
  int*   Mlds = (int*)(Alds + 2 * 16 * D_DIM);          // 1024 ints (4KB)
  float* RedV = (float*)(Mlds + TL);                    // 2*64 f32

  const int ch   = blockIdx.x;          // e-chunk 0..7
  const int g    = blockIdx.y;          // group 0..127
  const int tid  = threadIdx.x;         // 0..255
  const int wave = tid >> 5;
  const int et   = wave & 3;            // e-tile in chunk
  const int half = wave >> 2;           // 0/1 : t-tile phase
  const int lane = tid & 31;
  const int l15  = lane & 15;
  const int hh   = lane >> 4;

  { // W2 chunk -> LDS (rows e = ch*64 .. +63, all 512 d)
    const float4* src = (const float4*)(W2 + (size_t)ch * ECHUNK * D_DIM);
    float4* dst = (float4*)Wlds;
    for (int i = tid; i < ECHUNK * D_DIM / 4; i += blockDim.x) dst[i] = src[i];
  }
  { // mask row -> LDS
    const int4* src = (const int4*)(mask_total + (size_t)g * TL);
    int4* dst = (int4*)Mlds;
    for (int i = tid; i < TL / 4; i += blockDim.x) dst[i] = src[i];
  }
  __syncthreads();

  // B-fragment base: b[v] = W2[e0+n][k + 2*hh + v]
  const float* Bp = Wlds + ((et * 16 + l15) * D_DIM + 2 * hh);
  float* Amine = Alds + half * 16 * D_DIM;
  const float* Ap = Amine + (l15 * D_DIM + 2 * hh);
  const float4* seqg = (const float4*)(seq_total + (size_t)g * TL * D_DIM);
  const int tsub = tid & 127;           // thread id within half

  float vacc = 0.f;
  for (int tt = 0; tt < 32; ++tt) {
    const int t0 = tt * 32 + half * 16;
    __syncthreads();                     // previous tile consumed
    { // each half loads its own A tile: seq_total[g][t0..t0+15][:]
      const float4* src = seqg + (size_t)t0 * (D_DIM / 4);
      float4* dst = (float4*)Amine;
      for (int i = tsub; i < 16 * D_DIM / 4; i += 128) dst[i] = src[i];
    }
    __syncthreads();

    v8f acc = {0.f,0.f,0.f,0.f,0.f,0.f,0.f,0.f};
#pragma unroll 8
    for (int k = 0; k < D_DIM; k += 4) {
      v2f a = *(const v2f*)(Ap + k);
      v2f b = *(const v2f*)(Bp + k);
      acc = wmma_f32(a, b, acc);
    }
    // rows m = v + 8*hh -> t = t0+m ; column n = l15
#pragma unroll
    for (int v = 0; v < 8; ++v) {
      const int t = t0 + v + 8 * hh;
      const float x = lrelu(acc[v]);
      vacc += Mlds[t] ? x : 0.f;
    }
  }
  // combine hh halves (same column), then the two wave-halves, then store.
  vacc += __shfl_down(vacc, 16, 32);
  __syncthreads();
  if (hh == 0 && l15 == (lane & 15)) {   // lanes 0..15
    if (lane < 16) RedV[half * 64 + et * 16 + l15] = vacc;
  }
  __syncthreads();
  if (tid < 64)
    vsum[(size_t)g * D_DIM + ch * ECHUNK + tid] = RedV[tid] + RedV[64 + tid];
}

// -------- Kernel 2: partial[ch][g][c] = sum_{e in chunk} LReLU(<seq_cur[g][c], W1[e]>) * vsum[g][e]
__global__ void attn_kernel(const float* __restrict__ seq_cur,
                            const float* __restrict__ W1,
                            const float* __restrict__ vsum,
                            float*       __restrict__ partial) {
  extern __shared__ __align__(16) float smem[];
  float* Wlds = smem;                                   // 64*512
  float* Alds = Wlds + ECHUNK * D_DIM;                  // 2 * 16*512
  float* RedA = Alds + 2 * 16 * D_DIM;                  // 2*16 f32

  const int ch   = blockIdx.x;
  const int g    = blockIdx.y;
  const int tid  = threadIdx.x;
  const int wave = tid >> 5;
  const int et   = wave & 3;
  const int half = wave >> 2;
  const int lane = tid & 31;
  const int l15  = lane & 15;
  const int hh   = lane >> 4;
  const int tsub = tid & 127;

  {
    const float4* src = (const float4*)(W1 + (size_t)ch * ECHUNK * D_DIM);
    float4* dst = (float4*)Wlds;
    for (int i = tid; i < ECHUNK * D_DIM / 4; i += blockDim.x) dst[i] = src[i];
  }
  // per-lane column weight (constant over all c-tiles / K)
  const float vs = vsum[(size_t)g * D_DIM + ch * ECHUNK + et * 16 + l15];
  const float* Bp = Wlds + ((et * 16 + l15) * D_DIM + 2 * hh);
  float* Amine = Alds + half * 16 * D_DIM;
  const float* Ap = Amine + (l15 * D_DIM + 2 * hh);
  const float4* seqg = (const float4*)(seq_cur + (size_t)g * CL * D_DIM);
  float* myRed = RedA + half * 16;
  __syncthreads();

  for (int ct = 0; ct < 8; ++ct) {
    const int c0 = ct * 32 + half * 16;
    __syncthreads();
    {
      const float4* src = seqg + (size_t)c0 * (D_DIM / 4);
      float4* dst = (float4*)Amine;
      for (int i = tsub; i < 16 * D_DIM / 4; i += 128) dst[i] = src[i];
    }
    if (tsub < 16) myRed[tsub] = 0.f;
    __syncthreads();

    v8f acc = {0.f,0.f,0.f,0.f,0.f,0.f,0.f,0.f};
#pragma unroll 8
    for (int k = 0; k < D_DIM; k += 4) {
      v2f a = *(const v2f*)(Ap + k);
      v2f b = *(const v2f*)(Bp + k);
      acc = wmma_f32(a, b, acc);
    }
    // reduce over columns n (16 lanes of each hh-half); rows m = v + 8*hh
#pragma unroll
    for (int v = 0; v < 8; ++v) {
      float r = lrelu(acc[v]) * vs;
      r += __shfl_xor(r, 8, 32);
      r += __shfl_xor(r, 4, 32);
      r += __shfl_xor(r, 2, 32);
      r += __shfl_xor(r, 1, 32);
      if (l15 == 0) atomicAdd(&myRed[v + 8 * hh], r);   // across 4 waves of this half
    }
    __syncthreads();
    if (tsub < 16)
      partial[((size_t)ch * G_CNT + g) * CL + c0 + tsub] = myRed[tsub];
  }
}

// -------- Kernel 3: sum partials, scale, mask-fill, softmax over the 256 c's of each group.
__global__ void softmax_kernel(const float* __restrict__ partial,
                               const int*   __restrict__ mask_cur,
                               float*       __restrict__ out) {
  __shared__ float red[CL];
  const int g = blockIdx.x, c = threadIdx.x;
  float s = 0.f;
#pragma unroll
  for (int ch = 0; ch < 8; ++ch)
    s += partial[((size_t)ch * G_CNT + g) * CL + c];
  s *= 0.04419417382415922f;            // 1/sqrt(512)
  if (mask_cur[(size_t)g * CL + c] == 0) s = -1e9f;

  red[c] = s; __syncthreads();
  for (int off = CL / 2; off > 0; off >>= 1) {
    if (c < off) red[c] = fmaxf(red[c], red[c + off]);
    __syncthreads();
  }
  const float m = red[0]; __syncthreads();
  const float e = expf(s - m);
  red[c] = e; __syncthreads();
  for (int off = CL / 2; off > 0; off >>= 1) {
    if (c < off) red[c] += red[c + off];
    __syncthreads();
  }
  out[(size_t)g * CL + c] = e / red[0];
}

extern "C" void kernel_launch(void* const* d_in, const int* in_sizes, int n_in,
                              void* d_out, int out_size, void* d_ws, size_t ws_size,
                              hipStream_t stream) {
  (void)in_sizes; (void)n_in; (void)out_size; (void)ws_size;
  const float* seq_cur   = (const float*)d_in[0];
  const float* seq_total = (const float*)d_in[1];
  const int*   mask_cur  = (const int*)d_in[2];
  const int*   mask_tot  = (const int*)d_in[3];
  const float* W1        = (const float*)d_in[4];
  const float* W2        = (const float*)d_in[5];
  float* out = (float*)d_out;

  float* vsum    = (float*)d_ws;                 // 128*512 f32  (256 KB)
  float* partial = vsum + (size_t)G_CNT * D_DIM; // 8*128*256 f32 (1 MB)

  const size_t lds1 = (size_t)(ECHUNK * D_DIM + 2 * 16 * D_DIM) * sizeof(float)
                    + TL * sizeof(int) + 128 * sizeof(float);
  const size_t lds2 = (size_t)(ECHUNK * D_DIM + 2 * 16 * D_DIM + 32) * sizeof(float);
  hipFuncSetAttribute(reinterpret_cast<const void*>(vsum_kernel),
                      hipFuncAttributeMaxDynamicSharedMemorySize, (int)lds1);
  hipFuncSetAttribute(reinterpret_cast<const void*>(attn_kernel),
                      hipFuncAttributeMaxDynamicSharedMemorySize, (int)lds2);

  vsum_kernel<<<dim3(8, G_CNT), 256, lds1, stream>>>(seq_total, mask_tot, W2, vsum);
  attn_kernel<<<dim3(8, G_CNT), 256, lds2, stream>>>(seq_cur, W1, vsum, partial);
  softmax_kernel<<<G_CNT, CL, 0, stream>>>(partial, mask_cur, out);
}